// MyMultiheadAttention_77592879170150
// MI455X (gfx1250) — compile-verified
//
#include <hip/hip_runtime.h>
#include <math.h>

typedef __attribute__((ext_vector_type(2))) float v2f;
typedef __attribute__((ext_vector_type(8))) float v8f;
typedef __attribute__((ext_vector_type(4))) unsigned int u32x4;
typedef __attribute__((ext_vector_type(4))) int i32x4;
typedef __attribute__((ext_vector_type(8))) int i32x8;

#define D_MODEL   1024
#define NUM_HEADS 16
#define HEAD_DIM  64
#define SEQ       2048
#define BATCH     2
#define M_ROWS    (BATCH * SEQ)   // 4096 rows of X / attn

#define LDS_ROW   68              // 64 floats + 4 floats TDM pad -> bank-conflict-free

#if defined(__has_builtin)
#if __has_builtin(__builtin_amdgcn_tensor_load_to_lds)
#define HAVE_TDM 1
#endif
#endif

// Exact-fp32 CDNA5 WMMA: D(16x16,f32) = A(16x4,f32) * B(4x16,f32) + C
__device__ __forceinline__ v8f wmma_f32(v2f a, v2f b, v8f c) {
  return __builtin_amdgcn_wmma_f32_16x16x4_f32(
      /*neg_a=*/false, a, /*neg_b=*/false, b,
      /*c_mod=*/(short)0, c, /*reuse_a=*/false, /*reuse_b=*/false);
}

// ---------------------------------------------------------------------------
// TDM: DMA one 16x64-float tile (contiguous rows of a [S,64] tensor) into LDS
// with a 4-DWORD pad after every 64-DWORD row (row stride 68 floats in LDS).
// ---------------------------------------------------------------------------
#ifdef HAVE_TDM
__device__ __forceinline__ void tdm_load_tile(unsigned ldsOff, const float* gsrc)
{
  unsigned long long ga = (unsigned long long)(size_t)gsrc;
  u32x4 g0;
  g0[0] = 1u;                                         // count=1, user mode
  g0[1] = ldsOff;                                     // lds_addr (bytes)
  g0[2] = (unsigned)ga;                               // global_addr[31:0]
  g0[3] = (unsigned)((ga >> 32) & 0x1FFFFFFull)       // global_addr[56:32]
        | (2u << 30);                                 // type=2 (image)
  i32x8 g1;
  g1[0] = (2 << 16)        // data_size = 4 bytes
        | (1 << 20)        // pad_enable
        | (5 << 22)        // pad_interval: 64 DWORDs
        | (3 << 25);       // pad_amount:   4 DWORDs
  g1[1] = 64 << 16;        // tensor_dim0 = 64 elements
  g1[2] = 16 << 16;        // tensor_dim1 = 16 rows
  g1[3] = 64 << 16;        // tile_dim0   = 64
  g1[4] = 16;              // tile_dim1   = 16, tile_dim2 = 0
  g1[5] = 64;              // tensor_dim0_stride = 64 elements
  g1[6] = 0;
  g1[7] = 0;
  i32x4 gz = {0, 0, 0, 0};
#if defined(__clang_major__) && __clang_major__ >= 23
  i32x8 gz8 = {0, 0, 0, 0, 0, 0, 0, 0};
  __builtin_amdgcn_tensor_load_to_lds(g0, g1, gz, gz, gz8, 0);
#else
  __builtin_amdgcn_tensor_load_to_lds(g0, g1, gz, gz, 0);
#endif
}
#endif

// Issue the staging of one K/V tile into an LDS buffer (padded layout).
__device__ __forceinline__ void stage_issue(float* dst, const float* src, int tid)
{
#ifdef HAVE_TDM
  (void)dst;
  if (tid < 32) tdm_load_tile((unsigned)(size_t)dst, src);
#else
  // cooperative fallback: 1024 floats / 128 threads
#pragma unroll
  for (int i = 0; i < 8; ++i) {
    const int e = tid + i * 128;
    dst[(e >> 6) * LDS_ROW + (e & 63)] = src[e];
  }
#endif
}

__device__ __forceinline__ void stage_close(int tid)
{
#ifdef HAVE_TDM
  if (tid < 32) __builtin_amdgcn_s_wait_tensorcnt(0);
#endif
  __syncthreads();
}

// ---------------------------------------------------------------------------
// C[M,N] = X[M,K] * W[N,K]^T   (K = D_MODEL = 1024)
// One wave computes a 32x32 tile (2x2 of 16x16 WMMA accumulators).
// out_mode 0: row-major [M,N] store. out_mode 1: scatter to [B,H,S,64].
// ---------------------------------------------------------------------------
__global__ __launch_bounds__(256)
void gemm_xwT(const float* __restrict__ X, const float* __restrict__ W,
              float* __restrict__ out, int out_mode)
{
  const int lane  = threadIdx.x & 31;
  const int wave  = threadIdx.x >> 5;
  const int wtile = blockIdx.x * 8 + wave;       // 4096 wave-tiles total
  const int tn    = wtile & 31;                  // N/32 = 32
  const int tm    = wtile >> 5;                  // M/32 = 128
  const int ln    = lane & 15;
  const int half  = lane >> 4;

  const float* Abase = X + (size_t)(tm * 32 + ln) * D_MODEL + 2 * half;
  const float* Bbase = W + (size_t)(tn * 32 + ln) * D_MODEL + 2 * half;

  v8f acc[2][2];
  acc[0][0] = {}; acc[0][1] = {}; acc[1][0] = {}; acc[1][1] = {};

  for (int k0 = 0; k0 < D_MODEL; k0 += 16) {
    if (k0 + 64 < D_MODEL) {                     // stream-ahead hint
      __builtin_prefetch(Abase + k0 + 64);
      __builtin_prefetch(Bbase + k0 + 64);
    }
    v2f a0[4], a1[4], b0[4], b1[4];
#pragma unroll
    for (int kk = 0; kk < 4; ++kk) {
      const int c = k0 + kk * 4;
      a0[kk] = *(const v2f*)(Abase + c);
      a1[kk] = *(const v2f*)(Abase + 16 * D_MODEL + c);
      b0[kk] = *(const v2f*)(Bbase + c);
      b1[kk] = *(const v2f*)(Bbase + 16 * D_MODEL + c);
    }
#pragma unroll
    for (int kk = 0; kk < 4; ++kk) {
      acc[0][0] = wmma_f32(a0[kk], b0[kk], acc[0][0]);
      acc[0][1] = wmma_f32(a0[kk], b1[kk], acc[0][1]);
      acc[1][0] = wmma_f32(a1[kk], b0[kk], acc[1][0]);
      acc[1][1] = wmma_f32(a1[kk], b1[kk], acc[1][1]);
    }
  }

#pragma unroll
  for (int m2 = 0; m2 < 2; ++m2) {
#pragma unroll
    for (int n2 = 0; n2 < 2; ++n2) {
#pragma unroll
      for (int j = 0; j < 8; ++j) {
        const int row = tm * 32 + m2 * 16 + j + 8 * half;  // C-layout rows
        const int col = tn * 32 + n2 * 16 + ln;            // C-layout cols
        const float val = acc[m2][n2][j];
        if (out_mode == 0) {
          out[(size_t)row * D_MODEL + col] = val;
        } else {
          const int b = row >> 11, s = row & (SEQ - 1);
          const int h = col >> 6,  d = col & (HEAD_DIM - 1);
          out[(((size_t)(b * NUM_HEADS + h) * SEQ) + s) * HEAD_DIM + d] = val;
        }
      }
    }
  }
}

// ---------------------------------------------------------------------------
// RoPE on [B,H,S,64] (interleaved even/odd pairs). `scale` folds the 1/sqrt(d)
// attention score scale into Q (scale=0.125) and is 1.0 for K.
// ---------------------------------------------------------------------------
__global__ __launch_bounds__(256)
void rope_kernel(float* __restrict__ t, float scale)
{
  const int idx = blockIdx.x * blockDim.x + threadIdx.x;  // B*H*S*32 threads
  const int i  = idx & 31;                // rotation pair index
  const int s  = (idx >> 5) & (SEQ - 1);  // position
  const int bh = idx >> 16;               // b*H + h
  float* p = t + ((size_t)bh * SEQ + s) * HEAD_DIM + 2 * i;
  v2f x = *(v2f*)p;
  // inv_freq = 10000^(-i/32) = exp(-i * ln(10000)/32)
  const float inv_freq = __expf(-(float)i * 0.28782313662425f);
  const float ang = (float)s * inv_freq;
  const float sn = sinf(ang), cs = cosf(ang);
  v2f r;
  r.x = (x.x * cs - x.y * sn) * scale;
  r.y = (x.x * sn + x.y * cs) * scale;
  *(v2f*)p = r;
}

// ---------------------------------------------------------------------------
// Flash attention. One workgroup (4 waves) owns a 64-row query block; each
// wave owns 16 rows. K/V tiles (16x64) are TDM-DMA'd into double-buffered LDS
// (padded rows, bank-conflict-free) and shared by all 4 waves, with the next
// tile's DMA overlapped with compute on the current tile.
// ---------------------------------------------------------------------------
__global__ __launch_bounds__(128)
void flash_attn(const float* __restrict__ Q, const float* __restrict__ K,
                const float* __restrict__ V, float* __restrict__ O)
{
  __shared__ float ktile[2][16 * LDS_ROW];   // 2 x 4.25 KB
  __shared__ float vtile[2][16 * LDS_ROW];
  __shared__ float pstage[4][16 * 18];       // per-wave P staging

  const int tid  = threadIdx.x;
  const int lane = tid & 31;
  const int wave = tid >> 5;
  const int ln   = lane & 15;
  const int half = lane >> 4;
  const int bh   = blockIdx.y;               // b*16 + h
  const int qbase = blockIdx.x * 64;         // block's first query row
  const int qt0   = qbase + wave * 16;       // this wave's query tile

  const float* Qb = Q + ((size_t)bh * SEQ + qt0) * HEAD_DIM;
  const float* Kb = K + (size_t)bh * SEQ * HEAD_DIM;
  const float* Vb = V + (size_t)bh * SEQ * HEAD_DIM;
  float* shw = pstage[wave];

  // Q tile as 16 A-fragments (covers head dim 64 in K-chunks of 4)
  v2f qf[16];
#pragma unroll
  for (int kk = 0; kk < 16; ++kk)
    qf[kk] = *(const v2f*)(Qb + (size_t)ln * HEAD_DIM + kk * 4 + 2 * half);

  v8f o[4];
  o[0] = {}; o[1] = {}; o[2] = {}; o[3] = {};
  float rm[8], l[8];
#pragma unroll
  for (int j = 0; j < 8; ++j) { rm[j] = -__builtin_inff(); l[j] = 0.f; }

  const int kmax = qbase + 48;               // last (diagonal) tile of wave 3

  // ---- pipeline prologue: stage tile 0 ------------------------------------
  stage_issue(ktile[0], Kb, tid);
  stage_issue(vtile[0], Vb, tid);
  stage_close(tid);

  int buf = 0;
  for (int kt0 = 0; kt0 <= kmax; kt0 += 16) {
    // issue DMA for the next tile while computing on the current one
    if (kt0 + 16 <= kmax) {
      stage_issue(ktile[buf ^ 1], Kb + (size_t)(kt0 + 16) * HEAD_DIM, tid);
      stage_issue(vtile[buf ^ 1], Vb + (size_t)(kt0 + 16) * HEAD_DIM, tid);
    }

    if (kt0 <= qt0) {                        // causal: this wave has work
      const float* Kt = ktile[buf];
      const float* Vt = vtile[buf];

      // ---- scores: S = (Q * 1/8) @ K^T  (K fragments from LDS) ------------
      v8f sc = {};
#pragma unroll
      for (int kk = 0; kk < 16; ++kk) {
        v2f bk = *(const v2f*)(Kt + ln * LDS_ROW + kk * 4 + 2 * half);
        sc = wmma_f32(qf[kk], bk, sc);
      }
      if (kt0 == qt0) {                      // mask diagonal tile
#pragma unroll
        for (int j = 0; j < 8; ++j)
          if (ln > j + 8 * half) sc[j] = -__builtin_inff();
      }

      // ---- online softmax (row = one VGPR across a 16-lane half) ----------
      float p[8], sscale[8];
#pragma unroll
      for (int j = 0; j < 8; ++j) {
        float m = sc[j];
        m = fmaxf(m, __shfl_xor(m, 1, 32));
        m = fmaxf(m, __shfl_xor(m, 2, 32));
        m = fmaxf(m, __shfl_xor(m, 4, 32));
        m = fmaxf(m, __shfl_xor(m, 8, 32));
        const float mnew = fmaxf(rm[j], m);
        sscale[j] = __expf(rm[j] - mnew);
        rm[j] = mnew;
        const float pv = __expf(sc[j] - mnew);
        p[j] = pv;
        float rs = pv;
        rs += __shfl_xor(rs, 1, 32);
        rs += __shfl_xor(rs, 2, 32);
        rs += __shfl_xor(rs, 4, 32);
        rs += __shfl_xor(rs, 8, 32);
        l[j] = l[j] * sscale[j] + rs;
      }
#pragma unroll
      for (int g = 0; g < 4; ++g)
#pragma unroll
        for (int j = 0; j < 8; ++j)
          o[g][j] *= sscale[j];

      // ---- C-layout -> A-layout via per-wave LDS (same-wave DS in-order) --
#pragma unroll
      for (int j = 0; j < 8; ++j)
        shw[(j + 8 * half) * 18 + ln] = p[j];
      v2f pa[4];
#pragma unroll
      for (int kc = 0; kc < 4; ++kc)
        pa[kc] = *(const v2f*)(shw + ln * 18 + kc * 4 + 2 * half);

      // ---- O += P @ V  (V fragments from LDS, conflict-free banks) --------
#pragma unroll
      for (int g = 0; g < 4; ++g) {
#pragma unroll
        for (int kc = 0; kc < 4; ++kc) {
          const int kd = kc * 4 + 2 * half;
          v2f bv;
          bv.x = Vt[kd * LDS_ROW + g * 16 + ln];
          bv.y = Vt[(kd + 1) * LDS_ROW + g * 16 + ln];
          o[g] = wmma_f32(pa[kc], bv, o[g]);
        }
      }
    }

    // close pipeline stage: prefetch done + all waves done with current buf
    stage_close(tid);
    buf ^= 1;
  }

  // ---- normalize, store to [B,S,D] for the output projection --------------
  const int b = bh >> 4, h = bh & 15;
#pragma unroll
  for (int j = 0; j < 8; ++j) {
    const float inv = 1.0f / l[j];
    const int s = qt0 + j + 8 * half;
#pragma unroll
    for (int g = 0; g < 4; ++g)
      O[((size_t)(b * SEQ + s)) * D_MODEL + h * HEAD_DIM + g * 16 + ln]
          = o[g][j] * inv;
  }
}

// ---------------------------------------------------------------------------
extern "C" void kernel_launch(void* const* d_in, const int* in_sizes, int n_in,
                              void* d_out, int out_size, void* d_ws, size_t ws_size,
                              hipStream_t stream)
{
  const float* x  = (const float*)d_in[0];
  const float* Wq = (const float*)d_in[1];
  const float* Wk = (const float*)d_in[2];
  const float* Wv = (const float*)d_in[3];
  const float* Wo = (const float*)d_in[4];
  float* out = (float*)d_out;

  float* q   = (float*)d_ws;                         // [B,H,S,64]
  float* k   = q   + (size_t)M_ROWS * D_MODEL;       // [B,H,S,64]
  float* v   = k   + (size_t)M_ROWS * D_MODEL;       // [B,H,S,64]
  float* att = v   + (size_t)M_ROWS * D_MODEL;       // [B,S,D]

  // Q/K/V projections straight into head-major layout
  gemm_xwT<<<512, 256, 0, stream>>>(x, Wq, q, 1);
  gemm_xwT<<<512, 256, 0, stream>>>(x, Wk, k, 1);
  gemm_xwT<<<512, 256, 0, stream>>>(x, Wv, v, 1);

  // RoPE (score scale folded into Q)
  const int ropeThreads = BATCH * NUM_HEADS * SEQ * 32;   // 2,097,152
  rope_kernel<<<ropeThreads / 256, 256, 0, stream>>>(q, 0.125f);
  rope_kernel<<<ropeThreads / 256, 256, 0, stream>>>(k, 1.0f);

  // causal flash attention: 64 query rows per workgroup, TDM-staged K/V
  dim3 ag(SEQ / 64, BATCH * NUM_HEADS);                   // 32 x 32 blocks
  flash_attn<<<ag, 128, 0, stream>>>(q, k, v, att);

  // output projection
  gemm_xwT<<<512, 256, 0, stream>>>(att, Wo, out, 0);
}